// BPR_1726576855598
// MI455X (gfx1250) — compile-verified
//
#include <hip/hip_runtime.h>
#include <math.h>

typedef float v2f __attribute__((ext_vector_type(2)));
typedef float v8f __attribute__((ext_vector_type(8)));

#define U_NUM 100000
#define I_NUM 50000
#define EDGES 1600000
#define FACTOR 64
#define BATCH 4096

// ---------------- CSR build ----------------

__global__ void count_kernel(const int* __restrict__ eu, const int* __restrict__ ei,
                             int* __restrict__ cnt_u, int* __restrict__ cnt_i, int n) {
  int e = blockIdx.x * blockDim.x + threadIdx.x;
  if (e < n) {
    atomicAdd(&cnt_u[eu[e]], 1);
    atomicAdd(&cnt_i[ei[e]], 1);
  }
}

// single-block exclusive scan (chunks of 1024 with carried offset)
__global__ void scan_kernel(const int* __restrict__ cnt, int* __restrict__ off,
                            int* __restrict__ cur, int n) {
  __shared__ int sdata[1024];
  __shared__ int scarry;
  int t = threadIdx.x;
  if (t == 0) scarry = 0;
  __syncthreads();
  for (int base = 0; base < n; base += 1024) {
    int i = base + t;
    int x = (i < n) ? cnt[i] : 0;
    sdata[t] = x;
    __syncthreads();
    for (int d = 1; d < 1024; d <<= 1) {
      int v = (t >= d) ? sdata[t - d] : 0;
      __syncthreads();
      sdata[t] += v;
      __syncthreads();
    }
    int incl = sdata[t];
    int carry = scarry;
    if (i < n) { int ex = carry + incl - x; off[i] = ex; cur[i] = ex; }
    __syncthreads();
    if (t == 1023) scarry = carry + incl;
    __syncthreads();
  }
  if (t == 0) off[n] = scarry;
}

__global__ void scatter_kernel(const int* __restrict__ eu, const int* __restrict__ ei,
                               const float* __restrict__ ev,
                               int* __restrict__ cur_u, int* __restrict__ cur_i,
                               int* __restrict__ csr_u_idx, float* __restrict__ csr_u_val,
                               int* __restrict__ csr_i_idx, float* __restrict__ csr_i_val,
                               int n) {
  int e = blockIdx.x * blockDim.x + threadIdx.x;
  if (e < n) {
    int u = eu[e], it = ei[e];
    float v = ev[e];
    int pu = atomicAdd(&cur_u[u], 1);
    csr_u_idx[pu] = it; csr_u_val[pu] = v;
    int pi = atomicAdd(&cur_i[it], 1);
    csr_i_idx[pi] = u;  csr_i_val[pi] = v;
  }
}

// ---------------- SpMM: out[row] = dscale[row]*prev[row] + sum_e val*src[idx] ----------------
// one wave32 per row; each lane owns 2 consecutive floats of the 64-wide row.

__global__ void spmm_kernel(float* __restrict__ out, const float* __restrict__ src,
                            const float* __restrict__ prev, const float* __restrict__ dscale,
                            const int* __restrict__ off, const int* __restrict__ idx,
                            const float* __restrict__ val, int nrows) {
  int row = blockIdx.x * (blockDim.x >> 5) + (threadIdx.x >> 5);
  int lane = threadIdx.x & 31;
  if (row >= nrows) return;
  float d = dscale[row];
  const float* pr = prev + (size_t)row * FACTOR + 2 * lane;
  float ax = d * pr[0];
  float ay = d * pr[1];
  int s = off[row], e = off[row + 1];
  for (int p = s; p < e; ++p) {
    int c = idx[p];
    float v = val[p];
    const float2 r = *(const float2*)(src + (size_t)c * FACTOR + 2 * lane);
    ax = fmaf(v, r.x, ax);
    ay = fmaf(v, r.y, ay);
  }
  float2 res; res.x = ax; res.y = ay;
  *(float2*)(out + (size_t)row * FACTOR + 2 * lane) = res;
}

// ---------------- WMMA scoring ----------------
// 16 batch rows per wave. D = A x B with A[m][k]=ue[m][k0+k], B[k][n]=ie[n][k0+k];
// diag(D) gives the batched dot products. 32-bit A(16x4) and B(4x16) VGPR layouts
// coincide per-lane: lanes 0-15 hold K=k0,k0+1 of row (lane&15); lanes 16-31 hold
// K=k0+2,k0+3, so a single v2f load serves as both the A and B operand.

__device__ __forceinline__ v8f wmma_acc(v2f a, v2f b, v8f c) {
  return __builtin_amdgcn_wmma_f32_16x16x4_f32(false, a, false, b, (short)0, c,
                                               false, false);
}

__device__ __forceinline__ float diag_pick(v8f c, int j) {
  float a0 = (j & 1) ? c[1] : c[0];
  float a1 = (j & 1) ? c[3] : c[2];
  float a2 = (j & 1) ? c[5] : c[4];
  float a3 = (j & 1) ? c[7] : c[6];
  float b0 = (j & 2) ? a1 : a0;
  float b1 = (j & 2) ? a3 : a2;
  return (j & 4) ? b1 : b0;
}

__global__ void score_kernel(const float* __restrict__ u0, const float* __restrict__ u1,
                             const float* __restrict__ u2, const float* __restrict__ u3,
                             const float* __restrict__ i0, const float* __restrict__ i1,
                             const float* __restrict__ i2, const float* __restrict__ i3,
                             const long long* __restrict__ user,
                             const long long* __restrict__ item_i,
                             const long long* __restrict__ item_j,
                             float* __restrict__ out) {
  int wave = (blockIdx.x * blockDim.x + threadIdx.x) >> 5;
  int lane = threadIdx.x & 31;
  int g = wave * 16;
  if (g >= BATCH) return;

  int m = lane & 15;
  long long uid = user[g + m];
  long long ti  = item_i[g + m];
  long long tj  = item_j[g + m];
  int colsel = (lane >> 4) << 1;  // 0 for lanes 0-15, 2 for lanes 16-31

  const float* UT[4] = {u0, u1, u2, u3};
  const float* IT[4] = {i0, i1, i2, i3};

  v8f c_pi = {}, c_pj = {}, c_uu = {}, c_ii = {}, c_jj = {};
  for (int t = 0; t < 4; ++t) {
    const float* up = UT[t] + (size_t)uid * FACTOR + colsel;
    const float* pi = IT[t] + (size_t)ti  * FACTOR + colsel;
    const float* pj = IT[t] + (size_t)tj  * FACTOR + colsel;
#pragma unroll 4
    for (int c0 = 0; c0 < FACTOR; c0 += 4) {
      v2f au = *(const v2f*)(up + c0);
      v2f bi = *(const v2f*)(pi + c0);
      v2f bj = *(const v2f*)(pj + c0);
      c_pi = wmma_acc(au, bi, c_pi);
      c_pj = wmma_acc(au, bj, c_pj);
      c_uu = wmma_acc(au, au, c_uu);
      c_ii = wmma_acc(bi, bi, c_ii);
      c_jj = wmma_acc(bj, bj, c_jj);
    }
  }

  // diagonal owners: lanes 0-7 own m=lane (VGPR lane); lanes 24-31 own m=lane-16 (VGPR lane-24)
  bool owner = (lane < 8) || (lane >= 24);
  if (owner) {
    int j = lane & 7;
    float dpi = diag_pick(c_pi, j);
    float dpj = diag_pick(c_pj, j);
    float duu = diag_pick(c_uu, j);
    float dii = diag_pick(c_ii, j);
    float djj = diag_pick(c_jj, j);
    int mm = (lane < 8) ? j : (j + 8);
    int bb = g + mm;
    out[bb] = dpi;
    out[BATCH + bb] = dpj;
    float x = dpi - dpj;
    float ls = fminf(x, 0.0f) - log1pf(expf(-fabsf(x)));  // log_sigmoid(x)
    float l2 = 0.01f * (duu + dii + djj);
    const float invB = 1.0f / (float)BATCH;
    atomicAdd(&out[2 * BATCH + 0], (-ls + l2) * invB);  // loss
    atomicAdd(&out[2 * BATCH + 1], (-ls) * invB);       // loss2
  }
}

// ---------------- launch ----------------

extern "C" void kernel_launch(void* const* d_in, const int* in_sizes, int n_in,
                              void* d_out, int out_size, void* d_ws, size_t ws_size,
                              hipStream_t stream) {
  (void)in_sizes; (void)n_in; (void)out_size; (void)ws_size;
  const float* u0  = (const float*)d_in[0];
  const float* i0  = (const float*)d_in[1];
  const float* d_i = (const float*)d_in[2];
  const float* d_j = (const float*)d_in[3];
  const float* ev  = (const float*)d_in[4];
  const int*   eu  = (const int*)d_in[5];
  const int*   ei  = (const int*)d_in[6];
  const long long* user   = (const long long*)d_in[7];
  const long long* item_i = (const long long*)d_in[8];
  const long long* item_j = (const long long*)d_in[9];
  float* out = (float*)d_out;

  char* base = (char*)d_ws;
  size_t o = 0;
  auto take = [&](size_t bytes) -> char* {
    char* p = base + o;
    o = (o + bytes + 255) & ~(size_t)255;
    return p;
  };

  int*   cnt_u = (int*)take(sizeof(int) * U_NUM);
  int*   off_u = (int*)take(sizeof(int) * (U_NUM + 1));
  int*   cur_u = (int*)take(sizeof(int) * U_NUM);
  int*   cnt_i = (int*)take(sizeof(int) * I_NUM);
  int*   off_i = (int*)take(sizeof(int) * (I_NUM + 1));
  int*   cur_i = (int*)take(sizeof(int) * I_NUM);
  int*   csr_u_idx = (int*)take(sizeof(int) * EDGES);
  float* csr_u_val = (float*)take(sizeof(float) * EDGES);
  int*   csr_i_idx = (int*)take(sizeof(int) * EDGES);
  float* csr_i_val = (float*)take(sizeof(float) * EDGES);
  float* u1 = (float*)take(sizeof(float) * (size_t)U_NUM * FACTOR);
  float* u2 = (float*)take(sizeof(float) * (size_t)U_NUM * FACTOR);
  float* u3 = (float*)take(sizeof(float) * (size_t)U_NUM * FACTOR);
  float* i1 = (float*)take(sizeof(float) * (size_t)I_NUM * FACTOR);
  float* i2 = (float*)take(sizeof(float) * (size_t)I_NUM * FACTOR);
  float* i3 = (float*)take(sizeof(float) * (size_t)I_NUM * FACTOR);

  hipMemsetAsync(cnt_u, 0, sizeof(int) * U_NUM, stream);
  hipMemsetAsync(cnt_i, 0, sizeof(int) * I_NUM, stream);
  hipMemsetAsync((char*)d_out + sizeof(float) * 2 * BATCH, 0, 2 * sizeof(float), stream);

  const int EB = (EDGES + 255) / 256;
  count_kernel<<<EB, 256, 0, stream>>>(eu, ei, cnt_u, cnt_i, EDGES);
  scan_kernel<<<1, 1024, 0, stream>>>(cnt_u, off_u, cur_u, U_NUM);
  scan_kernel<<<1, 1024, 0, stream>>>(cnt_i, off_i, cur_i, I_NUM);
  scatter_kernel<<<EB, 256, 0, stream>>>(eu, ei, ev, cur_u, cur_i,
                                         csr_u_idx, csr_u_val, csr_i_idx, csr_i_val, EDGES);

  const int gu = (U_NUM + 7) / 8;  // 8 waves (rows) per 256-thread block
  const int gi = (I_NUM + 7) / 8;
  spmm_kernel<<<gu, 256, 0, stream>>>(u1, i0, u0, d_i, off_u, csr_u_idx, csr_u_val, U_NUM);
  spmm_kernel<<<gi, 256, 0, stream>>>(i1, u0, i0, d_j, off_i, csr_i_idx, csr_i_val, I_NUM);
  spmm_kernel<<<gu, 256, 0, stream>>>(u2, i1, u1, d_i, off_u, csr_u_idx, csr_u_val, U_NUM);
  spmm_kernel<<<gi, 256, 0, stream>>>(i2, u1, i1, d_j, off_i, csr_i_idx, csr_i_val, I_NUM);
  spmm_kernel<<<gu, 256, 0, stream>>>(u3, i2, u2, d_i, off_u, csr_u_idx, csr_u_val, U_NUM);
  spmm_kernel<<<gi, 256, 0, stream>>>(i3, u2, i2, d_j, off_i, csr_i_idx, csr_i_val, I_NUM);

  // BATCH/16 = 256 waves -> 8192 threads -> 32 blocks of 256
  score_kernel<<<(BATCH / 16) * 32 / 256, 256, 0, stream>>>(
      u0, u1, u2, u3, i0, i1, i2, i3, user, item_i, item_j, out);
}